// NeuralAggregation_10720238371128
// MI455X (gfx1250) — compile-verified
//
#include <hip/hip_runtime.h>

typedef __attribute__((ext_vector_type(2))) float v2f;
typedef __attribute__((ext_vector_type(8))) float v8f;

#define DIMK 256          // K = feature dim = output dim
#define WT_PITCH 260      // padded LDS row pitch (floats): 4*260 % 256B dodges bank alignment
#define ROWS_PER_WAVE 16
#define WAVES_PER_BLOCK 8
#define ROWS_PER_BLOCK (ROWS_PER_WAVE * WAVES_PER_BLOCK)

struct SharedBuf {
  float wt[DIMK * WT_PITCH];                    // W transposed: wt[n*PITCH + k] = W[k][n]
  float amax[WAVES_PER_BLOCK][ROWS_PER_WAVE];
  float amin[WAVES_PER_BLOCK][ROWS_PER_WAVE];
};

__global__ __launch_bounds__(256, 1)
void neural_agg_wmma_f32(const float* __restrict__ features,
                         const float* __restrict__ adjacency,
                         const float* __restrict__ W,
                         float* __restrict__ out,
                         int nNodes)
{
  __shared__ SharedBuf sb;

  const int tid = threadIdx.x;

  // ---- Stage W into LDS, transposed. Global reads coalesced row-major. ----
  for (int idx = tid; idx < DIMK * DIMK; idx += 256) {
    int k = idx >> 8;     // W row
    int n = idx & 255;    // W col
    sb.wt[n * WT_PITCH + k] = W[idx];
  }

  const int wave = tid >> 5;
  const int lane = tid & 31;     // wave32
  const int h = lane >> 4;       // lane half-select
  const int m = lane & 15;

  const int rowBase = (blockIdx.x * WAVES_PER_BLOCK + wave) * ROWS_PER_WAVE;

  // ---- Per-row adjacency max/min: lanes 0..15 each reduce one row of 32. ----
  if (lane < 16) {
    int r = rowBase + lane;
    int rc = r < nNodes ? r : nNodes - 1;
    const float4* ap = (const float4*)(adjacency + (size_t)rc * 32);
    float mx = -3.402823466e38f, mn = 3.402823466e38f;
    #pragma unroll
    for (int j = 0; j < 8; ++j) {
      float4 a = ap[j];
      mx = fmaxf(mx, fmaxf(fmaxf(a.x, a.y), fmaxf(a.z, a.w)));
      mn = fminf(mn, fminf(fminf(a.x, a.y), fminf(a.z, a.w)));
    }
    sb.amax[wave][lane] = mx;
    sb.amin[wave][lane] = mn;
  }

  __syncthreads();   // Wt (and reductions) visible

  // ---- GEMM: this wave computes a 16-row x 256-col slab, K = 256. ----
  // A fragment layout (V_WMMA_F32_16X16X4_F32): lane(h,m), VGPR j holds
  // features[rowA][k0 + 2h + j], j = 0,1 -> one global b64 per k-step.
  int rowA = rowBase + m;
  if (rowA >= nNodes) rowA = nNodes - 1;          // clamp tail loads; stores masked later
  const float* arow = features + (size_t)rowA * DIMK;

  v8f acc[16];
  #pragma unroll
  for (int nt = 0; nt < 16; ++nt) acc[nt] = (v8f)(0.0f);

  for (int k0 = 0; k0 < DIMK; k0 += 4) {
    v2f afrag = *(const v2f*)(arow + k0 + 2 * h);
    #pragma unroll
    for (int nt = 0; nt < 16; ++nt) {
      // B fragment: VGPR j holds W[k0 + 2h + j][nt*16 + m] -> contiguous in wt
      const float* bp = &sb.wt[(nt * 16 + m) * WT_PITCH + (k0 + 2 * h)];
      v2f bfrag = *(const v2f*)bp;                 // ds_load_b64
      acc[nt] = __builtin_amdgcn_wmma_f32_16x16x4_f32(
          false, afrag, false, bfrag, (short)0, acc[nt], false, false);
    }
  }

  // ---- Epilogue: out = max(0, max(amax*p, amin*p)). ----
  // C/D layout: acc[nt] VGPR v, lane(h,m) = element (row 8h+v, col nt*16+m).
  float amx[8], amn[8];
  #pragma unroll
  for (int v = 0; v < 8; ++v) {
    amx[v] = sb.amax[wave][8 * h + v];
    amn[v] = sb.amin[wave][8 * h + v];
  }

  #pragma unroll
  for (int nt = 0; nt < 16; ++nt) {
    int col = nt * 16 + m;
    #pragma unroll
    for (int v = 0; v < 8; ++v) {
      int row = rowBase + 8 * h + v;
      if (row < nNodes) {
        float p = acc[nt][v];
        float val = fmaxf(fmaxf(amx[v] * p, amn[v] * p), 0.0f);
        out[(size_t)row * DIMK + col] = val;
      }
    }
  }
}

extern "C" void kernel_launch(void* const* d_in, const int* in_sizes, int n_in,
                              void* d_out, int out_size, void* d_ws, size_t ws_size,
                              hipStream_t stream) {
  const float* features  = (const float*)d_in[0];   // [N, 256] f32
  const float* adjacency = (const float*)d_in[1];   // [N, 32]  f32
  const float* W         = (const float*)d_in[2];   // [256, 256] f32
  float* out = (float*)d_out;                       // [N, 256] f32

  const int nNodes = in_sizes[0] / DIMK;
  const int blocks = (nNodes + ROWS_PER_BLOCK - 1) / ROWS_PER_BLOCK;
  neural_agg_wmma_f32<<<blocks, 256, 0, stream>>>(features, adjacency, W, out, nNodes);
}